// HungarianMatcher_40200893890946
// MI455X (gfx1250) — compile-verified
//
#include <hip/hip_runtime.h>
#include <hip/hip_bf16.h>
#include <math.h>

typedef __attribute__((ext_vector_type(2))) float v2f;
typedef __attribute__((ext_vector_type(8))) float v8f;

#define BATCH 32
#define NP 256   // number of predictions (rows of C)
#define NM 256   // number of targets (cols of C)

// ---------------------------------------------------------------------------
// Kernel 1: pairwise L2 cost matrix via V_WMMA_F32_16X16X4_F32.
// dist^2(n,m) = [px,py,px^2+py^2,1] . [-2tx,-2ty,1,tx^2+ty^2]  (K = 4)
// One wave computes one 16x16 tile of C; 8 waves per 256-thread block.
// Final sqrt uses the hardware v_sqrt_f32 (TRANS unit, co-issues with WMMA);
// ~1 ulp is ample for comparing assignment costs of points in [0,1).
// ---------------------------------------------------------------------------
__global__ __launch_bounds__(256) void cost_wmma_kernel(
    const float* __restrict__ pred, const float* __restrict__ tgt,
    float* __restrict__ C) {
  const int wave = blockIdx.x * (blockDim.x >> 5) + (threadIdx.x >> 5);
  const int lane = threadIdx.x & 31;
  const int b    = wave >> 8;       // 16*16 = 256 tiles per batch
  const int rem  = wave & 255;
  const int tn   = rem >> 4;        // row-tile
  const int tm   = rem & 15;        // col-tile

  const int  r16 = lane & 15;
  const bool hi  = lane >= 16;

  // A fragment (16x4 f32): lanes 0-15 hold K=0,1 ; lanes 16-31 hold K=2,3
  const int arow = tn * 16 + r16;
  const float px = pred[(b * NP + arow) * 2 + 0];
  const float py = pred[(b * NP + arow) * 2 + 1];
  v2f a;
  if (!hi) { a.x = px;                a.y = py;   }
  else     { a.x = px * px + py * py; a.y = 1.0f; }

  // B fragment (4x16 f32): lanes 0-15 hold K=0,1 ; lanes 16-31 hold K=2,3
  const int bcol = tm * 16 + r16;
  const float tx = tgt[(b * NM + bcol) * 2 + 0];
  const float ty = tgt[(b * NM + bcol) * 2 + 1];
  v2f bf;
  if (!hi) { bf.x = -2.0f * tx; bf.y = -2.0f * ty;        }
  else     { bf.x = 1.0f;       bf.y = tx * tx + ty * ty; }

  v8f c = {};
  v8f d = __builtin_amdgcn_wmma_f32_16x16x4_f32(
      /*neg_a=*/false, a, /*neg_b=*/false, bf,
      /*c_mod=*/(short)0, c, /*reuse_a=*/false, /*reuse_b=*/false);

  // C/D layout: VGPR r -> (M = r [+8 for lanes 16-31], N = lane&15)
  const int colo  = tm * 16 + r16;
  const int rbase = tn * 16 + (hi ? 8 : 0);
#pragma unroll
  for (int r = 0; r < 8; ++r) {
    const float sq = fmaxf(d[r], 0.0f);
    C[(b * NP + (rbase + r)) * NM + colo] = __builtin_amdgcn_sqrtf(sq);
  }
}

// ---------------------------------------------------------------------------
// Kernel 2: Jonker-Volgenant shortest-augmenting-path LSA.
// One 256-thread workgroup per batch; thread t owns column t+1.
// If k = tgt_num[b] < 256 the problem is solved transposed (rows = targets),
// matching the reference. All solver state lives in LDS.
// ---------------------------------------------------------------------------
__global__ __launch_bounds__(256) void jv_kernel(
    const float* __restrict__ C, const int* __restrict__ tnum,
    float* __restrict__ out_ids, float* __restrict__ tgt_ids,
    float* __restrict__ h_cost) {
  __shared__ float u[NP + 1];
  __shared__ float v[NM + 1];
  __shared__ float minv[NM + 1];
  __shared__ int   p[NM + 1];
  __shared__ int   way[NM + 1];
  __shared__ int   used[NM + 1];
  __shared__ float rv[256];
  __shared__ int   ri[256];
  __shared__ int   t_of_p[NP];
  __shared__ int   s_j0;

  const int b   = blockIdx.x;
  const int tid = threadIdx.x;
  const int col = tid + 1;
  const float INF = 3.0e38f;

  int k = tnum[b];
  if (k < 1)  k = NM;
  if (k > NM) k = NM;
  const bool transposed = (k < NP);  // ref transposes when rows > cols
  const int  n_rows = k;             // rows of the (possibly transposed) cost
  const float* Cb = C + (size_t)b * NP * NM;

  u[tid] = 0.0f; v[tid] = 0.0f; p[tid] = 0; way[tid] = 0;
  if (tid == 0) { u[NP] = 0.0f; v[NM] = 0.0f; p[NM] = 0; way[NM] = 0; }
  t_of_p[tid] = -1;
  __syncthreads();

  for (int i = 1; i <= n_rows; ++i) {
    if (tid == 0) { p[0] = i; s_j0 = 0; minv[0] = INF; used[0] = 0; }
    minv[col] = INF;
    used[col] = 0;
    __syncthreads();

    while (true) {
      if (tid == 0) used[s_j0] = 1;
      __syncthreads();
      const int j0 = s_j0;
      const int i0 = p[j0];

      float mval = INF;
      if (!used[col]) {
        const int ci = i0 - 1;
        const int cj = col - 1;
        const float cval = transposed ? Cb[cj * NM + ci] : Cb[ci * NM + cj];
        const float cur  = cval - u[i0] - v[col];
        if (cur < minv[col]) { minv[col] = cur; way[col] = j0; }
        mval = minv[col];
      }
      rv[tid] = mval; ri[tid] = col;
      __syncthreads();
      // Block argmin with lowest-index tie-break (matches np.argmin).
      for (int s = 128; s > 0; s >>= 1) {
        if (tid < s) {
          const float ov = rv[tid + s];
          const int   oi = ri[tid + s];
          if (ov < rv[tid] || (ov == rv[tid] && oi < ri[tid])) {
            rv[tid] = ov; ri[tid] = oi;
          }
        }
        __syncthreads();
      }
      const float delta = rv[0];
      const int   j1    = ri[0];

      if (used[col]) { v[col] -= delta; u[p[col]] += delta; }  // distinct rows
      else           { minv[col] -= delta; }
      if (tid == 0)  { v[0] -= delta; u[p[0]] += delta; s_j0 = j1; }
      __syncthreads();

      if (p[j1] == 0) break;
    }

    if (tid == 0) {  // augment along the alternating path
      int j0a = s_j0;
      while (j0a) { const int jn = way[j0a]; p[j0a] = p[jn]; j0a = jn; }
    }
    __syncthreads();
  }

  // Extract matching as t_of_p[pred] = target.
  if (p[col] > 0) {
    if (transposed) t_of_p[col - 1] = p[col] - 1;   // rows were targets
    else            t_of_p[p[col] - 1] = col - 1;   // rows were preds
  }
  __syncthreads();

  if (tid == 0) {  // compact in ascending pred order (ref: argsort(col))
    int cnt = 0;
    float hc = 0.0f;
    for (int pi = 0; pi < NP; ++pi) {
      const int t = t_of_p[pi];
      if (t >= 0) {
        out_ids[b * NP + cnt] = (float)pi;
        tgt_ids[b * NP + cnt] = (float)t;
        hc += Cb[pi * NM + t];
        ++cnt;
      }
    }
    for (; cnt < NP; ++cnt) {
      out_ids[b * NP + cnt] = -1.0f;
      tgt_ids[b * NP + cnt] = -1.0f;
    }
    h_cost[b] = hc;
  }
}

extern "C" void kernel_launch(void* const* d_in, const int* in_sizes, int n_in,
                              void* d_out, int out_size, void* d_ws, size_t ws_size,
                              hipStream_t stream) {
  (void)in_sizes; (void)n_in; (void)out_size; (void)ws_size;
  const float* pred = (const float*)d_in[0];
  const float* tgt  = (const float*)d_in[1];
  const int*   tnum = (const int*)d_in[2];
  // d_in[3] = l_type (scalar, always 2 per setup) -> hard-coded L2.

  float* C       = (float*)d_ws;                 // [B][NP][NM] cost, 8 MiB
  float* out_ids = (float*)d_out;                // [B][NP]
  float* tgt_ids = out_ids + BATCH * NP;         // [B][NP]
  float* h_cost  = tgt_ids + BATCH * NP;         // [B]

  // 8192 tiles, 8 waves (tiles) per 256-thread block.
  cost_wmma_kernel<<<(BATCH * 16 * 16) / 8, 256, 0, stream>>>(pred, tgt, C);
  jv_kernel<<<BATCH, 256, 0, stream>>>(C, tnum, out_ids, tgt_ids, h_cost);
}